// LabelAttention2_13348758356159
// MI455X (gfx1250) — compile-verified
//
#include <hip/hip_runtime.h>
#include <hip/hip_bf16.h>
#include <math.h>

// ---------------------------------------------------------------------------
// Shapes: B=8, S=4096, L=64, D=1024, H=256.
// Collapsed math (see analysis): scores[b,s] = (Q[b,s,:].w + c)/16 - 1e9*mask,
// attn = softmax_s(scores), context[b,:] = sum_s attn[b,s]*V[b,s,:].
// Memory floor: read Q (128MB) + V (128MB) -> ~11us @ 23.3 TB/s. All kernels
// are streaming/bandwidth-bound; WMMA (fp32 16x16x4) carries the Q-dot path.
// ---------------------------------------------------------------------------

typedef __attribute__((ext_vector_type(2))) float v2f;
typedef __attribute__((ext_vector_type(8))) float v8f;

#define NB 8
#define NS 4096
#define ND 1024
#define NH 256
#define NL 64

// ---------------------------------------------------------------------------
// Kernel A: prepass. ks = colsum(K); ksum = Wk_w*ks + 64*Wk_b;
// w = Wq_w^T * ksum; c = Wq_b . ksum.  Also zeroes the context output region.
// One workgroup, 256 threads. ~2.3 MB of reads total -> negligible.
// ---------------------------------------------------------------------------
__global__ __launch_bounds__(256) void la_prepass(
    const float* __restrict__ K, const float* __restrict__ Wk_w,
    const float* __restrict__ Wk_b, const float* __restrict__ Wq_w,
    const float* __restrict__ Wq_b, float* __restrict__ wOut,
    float* __restrict__ cOut, float* __restrict__ ctxZero) {
  __shared__ float ks[ND];
  __shared__ float ksum[NH];
  __shared__ float red[256];
  const int tid = threadIdx.x;

  // zero context accumulation region of d_out (runs before kernel D in-stream)
  for (int i = tid; i < NB * ND; i += 256) ctxZero[i] = 0.0f;

  // ks[d] = sum_l K[l,d]   (coalesced across threads)
  for (int d = tid; d < ND; d += 256) {
    float s = 0.0f;
    for (int l = 0; l < NL; ++l) s += K[l * ND + d];
    ks[d] = s;
  }
  __syncthreads();

  // ksum[h] = ks . Wk_w[h,:] + 64*Wk_b[h]   (one head per thread)
  {
    const float* row = Wk_w + (size_t)tid * ND;
    float s = 0.0f;
    for (int d = 0; d < ND; ++d) s += ks[d] * row[d];
    ksum[tid] = s + 64.0f * Wk_b[tid];
  }
  __syncthreads();

  // w[d] = sum_h ksum[h] * Wq_w[h,d]   (coalesced across threads)
  for (int d = tid; d < ND; d += 256) {
    float s = 0.0f;
    for (int h = 0; h < NH; ++h) s += ksum[h] * Wq_w[(size_t)h * ND + d];
    wOut[d] = s;
  }

  // c = Wq_b . ksum
  red[tid] = Wq_b[tid] * ksum[tid];
  __syncthreads();
  for (int off = 128; off > 0; off >>= 1) {
    if (tid < off) red[tid] += red[tid + off];
    __syncthreads();
  }
  if (tid == 0) *cOut = red[0];
}

// ---------------------------------------------------------------------------
// Kernel B: scores via V_WMMA_F32_16X16X4_F32.
// Each wave owns 16 flattened rows of Q (row = b*S+s; 4096%16==0 so a tile
// never crosses a batch). A-tile = Q[16 rows, k..k+3] in the ISA 16x4 f32
// layout (lane m<16: K0/K1, lane m>=16: K2/K3). B-tile has every column equal
// to the w chunk, so every C column equals the 16 scores; lanes 0 and 16 hold
// rows 0-7 / 8-15 in their 8 accumulator VGPRs per the C/D layout.
// Grid: 256 blocks x 256 thr (8 waves) x 16 rows = 32768 rows exactly.
// ---------------------------------------------------------------------------
__global__ __launch_bounds__(256) void la_scores(
    const float* __restrict__ Q, const unsigned char* __restrict__ mask,
    const float* __restrict__ wVec, const float* __restrict__ cPtr,
    float* __restrict__ rawOut) {
  __shared__ float lw[ND];
  const int tid = threadIdx.x;
  for (int i = tid; i < ND; i += 256) lw[i] = wVec[i];
  __syncthreads();

  const int lane = tid & 31;
  const int wave = tid >> 5;
  const int rowbase = (blockIdx.x * 8 + wave) * 16;
  const float* qrow = Q + (size_t)(rowbase + (lane & 15)) * ND;
  const int koff = (lane >> 4) << 1;  // 0 for lanes 0-15, 2 for lanes 16-31

  v8f acc = {};
  for (int k = 0; k < ND; k += 4) {
    if ((k & 127) == 0 && k < 512)
      __builtin_prefetch(qrow + k + 512, 0, 0);  // global_prefetch_b8
    v2f a = *(const v2f*)(qrow + k + koff);      // A: global_load_b64
    v2f b = *(const v2f*)(&lw[k + koff]);        // B: ds_load_b64
    acc = __builtin_amdgcn_wmma_f32_16x16x4_f32(
        /*neg_a=*/false, a, /*neg_b=*/false, b,
        /*c_mod=*/(short)0, acc, /*reuse_a=*/false, /*reuse_b=*/false);
  }

  const float c = *cPtr;
  if (lane == 0 || lane == 16) {
    const int mbase = rowbase + ((lane >> 4) << 3);  // rows 0-7 or 8-15
    for (int r = 0; r < 8; ++r) {
      const int row = mbase + r;
      float raw = (acc[r] + c) * 0.0625f;  // / sqrt(256)
      if (mask[row]) raw -= 1e9f;
      rawOut[row] = raw;
    }
  }
}

// ---------------------------------------------------------------------------
// Kernel C: per-batch softmax over S=4096, in place in the attn output region.
// 8 blocks x 512 threads; scores staged in LDS (16KB).
// ---------------------------------------------------------------------------
__global__ __launch_bounds__(512) void la_softmax(float* __restrict__ attn) {
  __shared__ float buf[NS];
  __shared__ float red[512];
  const int b = blockIdx.x;
  const int tid = threadIdx.x;
  float* p = attn + (size_t)b * NS;

  float lmax = -3.4e38f;
  for (int i = tid; i < NS; i += 512) {
    float v = p[i];
    buf[i] = v;
    lmax = fmaxf(lmax, v);
  }
  red[tid] = lmax;
  __syncthreads();
  for (int off = 256; off > 0; off >>= 1) {
    if (tid < off) red[tid] = fmaxf(red[tid], red[tid + off]);
    __syncthreads();
  }
  const float m = red[0];
  __syncthreads();

  float lsum = 0.0f;
  for (int i = tid; i < NS; i += 512) {
    float e = expf(buf[i] - m);
    buf[i] = e;
    lsum += e;
  }
  red[tid] = lsum;
  __syncthreads();
  for (int off = 256; off > 0; off >>= 1) {
    if (tid < off) red[tid] += red[tid + off];
    __syncthreads();
  }
  const float inv = 1.0f / red[0];
  for (int i = tid; i < NS; i += 512) p[i] = buf[i] * inv;
}

// ---------------------------------------------------------------------------
// Kernel D: context[b,:] += sum_s attn[b,s] * V[b,s,:].
// Grid: 8 batches x 32 s-chunks (128 rows each) = 256 blocks x 256 threads.
// V reads fully coalesced (1024 consecutive floats per s); partial sums
// merged with global_atomic_add_f32 into the pre-zeroed context region.
// ---------------------------------------------------------------------------
__global__ __launch_bounds__(256) void la_context(
    const float* __restrict__ V, const float* __restrict__ attn,
    float* __restrict__ ctx) {
  __shared__ float la[128];
  const int b = blockIdx.x >> 5;
  const int sc = blockIdx.x & 31;
  const int tid = threadIdx.x;
  const int s0 = sc * 128;

  if (tid < 128) la[tid] = attn[(size_t)b * NS + s0 + tid];
  __syncthreads();

  float a0 = 0.0f, a1 = 0.0f, a2 = 0.0f, a3 = 0.0f;
  const float* vb = V + ((size_t)b * NS + s0) * ND;
  for (int s = 0; s < 128; ++s) {
    const float w = la[s];
    const float* vr = vb + (size_t)s * ND;
    if ((s & 7) == 0) __builtin_prefetch(vr + 8 * ND + tid, 0, 0);
    a0 += w * vr[tid];
    a1 += w * vr[tid + 256];
    a2 += w * vr[tid + 512];
    a3 += w * vr[tid + 768];
  }
  float* cb = ctx + (size_t)b * ND;
  atomicAdd(&cb[tid], a0);
  atomicAdd(&cb[tid + 256], a1);
  atomicAdd(&cb[tid + 512], a2);
  atomicAdd(&cb[tid + 768], a3);
}

// ---------------------------------------------------------------------------
// Launch. Inputs: 0=Q 1=K 2=V 3=attn_mask(bool,1B) 4=Wq_w 5=Wq_b 6=Wk_w 7=Wk_b
// d_out: [context (8*1024) | attn (8*4096)].  ws: [w (1024) | c (1)].
// ---------------------------------------------------------------------------
extern "C" void kernel_launch(void* const* d_in, const int* in_sizes, int n_in,
                              void* d_out, int out_size, void* d_ws,
                              size_t ws_size, hipStream_t stream) {
  const float* Q = (const float*)d_in[0];
  const float* K = (const float*)d_in[1];
  const float* V = (const float*)d_in[2];
  const unsigned char* mask = (const unsigned char*)d_in[3];
  const float* Wq_w = (const float*)d_in[4];
  const float* Wq_b = (const float*)d_in[5];
  const float* Wk_w = (const float*)d_in[6];
  const float* Wk_b = (const float*)d_in[7];

  float* out = (float*)d_out;
  float* ctx = out;              // 8*1024
  float* attn = out + NB * ND;   // 8*4096 (raw scores, then softmaxed in place)

  float* wVec = (float*)d_ws;    // 1024 floats
  float* cPtr = wVec + ND;       // 1 float

  la_prepass<<<1, 256, 0, stream>>>(K, Wk_w, Wk_b, Wq_w, Wq_b, wVec, cPtr, ctx);
  la_scores<<<256, 256, 0, stream>>>(Q, mask, wVec, cPtr, attn);
  la_softmax<<<NB, 512, 0, stream>>>(attn);
  la_context<<<256, 256, 0, stream>>>(V, attn, ctx);
}